// down_feature_48309792145533
// MI455X (gfx1250) — compile-verified
//
#include <hip/hip_runtime.h>
#include <cstdint>
#include <cstddef>

#define NB      16
#define CY      512
#define CX      3
#define NPTS    16384
#define KDS     4096

#define TILE_COLS   1024
#define CHUNK_ROWS  8
#define NCHUNKS     (CY / CHUNK_ROWS)   // 64

// ---------------- CDNA5 async global->LDS helpers (gfx1250) ----------------
// Each lane copies 16 bytes from its global address to its LDS address.
// Tracked with ASYNCcnt (per-wave); enforced with s_wait_asynccnt.
__device__ __forceinline__ void async_load_b128(uint32_t lds_addr, const void* gaddr) {
    asm volatile("global_load_async_to_lds_b128 %0, %1, off"
                 :: "v"(lds_addr), "v"((uint64_t)(uintptr_t)gaddr)
                 : "memory");
}
__device__ __forceinline__ void wait_async_le(int) {} // (doc helper; real ones below)
__device__ __forceinline__ void wait_async_le8() {
    asm volatile("s_wait_asynccnt 8" ::: "memory");
}
__device__ __forceinline__ void wait_async_0() {
    asm volatile("s_wait_asynccnt 0" ::: "memory");
}

// ---------------- Kernel 1: z[b][i] = max_c y[b][c][i] ----------------
// Streams y (512 MiB) through LDS with double-buffered async copies.
// Lane-private LDS slots: each lane reads back exactly the 16B it loaded,
// so only per-wave ASYNCcnt waits are needed (no workgroup barriers).
__global__ __launch_bounds__(256)
void colmax_kernel(const float* __restrict__ y, float* __restrict__ z) {
    __shared__ float buf[2][CHUNK_ROWS][TILE_COLS];   // 64 KiB
    const int b    = blockIdx.y;
    const int col0 = blockIdx.x * TILE_COLS;
    const int tid  = threadIdx.x;

    const float* ybase = y + ((size_t)b * CY) * NPTS + col0 + tid * 4;
    float* pA = &buf[0][0][tid * 4];
    float* pB = &buf[1][0][tid * 4];
    // low 32 bits of a flat pointer into LDS == wave-relative LDS byte address
    const uint32_t ldsA = (uint32_t)(uintptr_t)pA;
    const uint32_t ldsB = (uint32_t)(uintptr_t)pB;

    // prime: chunk 0 -> buffer A
#pragma unroll
    for (int r = 0; r < CHUNK_ROWS; ++r)
        async_load_b128(ldsA + (uint32_t)(r * TILE_COLS * 4),
                        ybase + (size_t)r * NPTS);

    float4 m;
    m.x = m.y = m.z = m.w = -__builtin_inff();

    for (int chunk = 0; chunk < NCHUNKS; ++chunk) {
        const float* cur = (chunk & 1) ? pB : pA;
        if (chunk + 1 < NCHUNKS) {
            // issue next chunk into the other buffer (its prior contents were
            // fully consumed in the previous iteration; DS reads already completed
            // because their values fed the VALU max below)
            const uint32_t nxt = (chunk & 1) ? ldsA : ldsB;
            const float* g = ybase + (size_t)(chunk + 1) * CHUNK_ROWS * NPTS;
#pragma unroll
            for (int r = 0; r < CHUNK_ROWS; ++r)
                async_load_b128(nxt + (uint32_t)(r * TILE_COLS * 4),
                                g + (size_t)r * NPTS);
            wait_async_le8();   // older 8 (current buffer) complete
        } else {
            wait_async_0();
        }
#pragma unroll
        for (int r = 0; r < CHUNK_ROWS; ++r) {
            const float4 v = *(const float4*)(cur + (size_t)r * TILE_COLS);
            m.x = fmaxf(m.x, v.x);
            m.y = fmaxf(m.y, v.y);
            m.z = fmaxf(m.z, v.z);
            m.w = fmaxf(m.w, v.w);
        }
    }

    float* zp = z + (size_t)b * NPTS + col0 + tid * 4;
    *(float4*)zp = m;
}

// ---------------- Kernel 2: full descending bitonic sort per batch ----------
// key = (ordered_u32(value) << 32) | (16383 - index): descending sort on key
// == descending by value with ascending-index tiebreak (== lax.top_k order).
__global__ __launch_bounds__(1024)
void topk_sort_kernel(const float* __restrict__ z, int* __restrict__ idx) {
    extern __shared__ unsigned long long keys[];   // 16384 * 8B = 128 KiB dynamic LDS
    const int b   = blockIdx.x;
    const int tid = threadIdx.x;

    for (int i = tid; i < NPTS; i += 1024) {
        uint32_t u = __float_as_uint(z[(size_t)b * NPTS + i]);
        u = (u & 0x80000000u) ? ~u : (u | 0x80000000u);   // order-preserving map
        keys[i] = ((unsigned long long)u << 32) | (uint32_t)(NPTS - 1 - i);
    }
    __syncthreads();

    for (unsigned k = 2; k <= (unsigned)NPTS; k <<= 1) {
        for (unsigned j = k >> 1; j > 0; j >>= 1) {
            for (unsigned i = tid; i < (unsigned)NPTS; i += 1024) {
                const unsigned ixj = i ^ j;
                if (ixj > i) {
                    const unsigned long long a = keys[i];
                    const unsigned long long c = keys[ixj];
                    const bool desc = ((i & k) == 0);      // overall descending
                    if (desc ? (a < c) : (a > c)) {
                        keys[i]   = c;
                        keys[ixj] = a;
                    }
                }
            }
            __syncthreads();
        }
    }

    for (int j = tid; j < KDS; j += 1024)
        idx[(size_t)b * KDS + j] = (NPTS - 1) - (int)(keys[j] & 0xFFFFFFFFull);
}

// ---------------- Kernel 3: gather x and y columns at top-k indices ---------
// grid.x in [0,64]: 0..63 -> 8 y-channels each; 64 -> the 3 x-channels.
__global__ __launch_bounds__(256)
void gather_kernel(const float* __restrict__ x, const float* __restrict__ y,
                   const int* __restrict__ idx, float* __restrict__ out) {
    __shared__ int sidx[KDS];   // 16 KiB
    const int b   = blockIdx.y;
    const int g   = blockIdx.x;
    const int tid = threadIdx.x;

    for (int j = tid; j < KDS; j += 256)
        sidx[j] = idx[(size_t)b * KDS + j];
    __syncthreads();

    if (g < CY / 8) {
        const float* yb   = y   + ((size_t)b * CY) * NPTS;
        float*       optr = out + (size_t)NB * CX * KDS + ((size_t)b * CY) * KDS;
#pragma unroll
        for (int cc = 0; cc < 8; ++cc) {
            const int c = g * 8 + cc;
            const float* row  = yb   + (size_t)c * NPTS;
            float*       orow = optr + (size_t)c * KDS;
            for (int j = tid; j < KDS; j += 256)
                orow[j] = row[sidx[j]];
        }
    } else {
        const float* xb   = x   + (size_t)b * CX * NPTS;
        float*       optr = out + (size_t)b * CX * KDS;
#pragma unroll
        for (int c = 0; c < CX; ++c)
            for (int j = tid; j < KDS; j += 256)
                optr[(size_t)c * KDS + j] = xb[(size_t)c * NPTS + sidx[j]];
    }
}

// ---------------- Launch ----------------
extern "C" void kernel_launch(void* const* d_in, const int* in_sizes, int n_in,
                              void* d_out, int out_size, void* d_ws, size_t ws_size,
                              hipStream_t stream) {
    const float* x = (const float*)d_in[0];   // (16, 3, 16384) f32
    const float* y = (const float*)d_in[1];   // (16, 512, 16384) f32
    float* out = (float*)d_out;               // [xyz | pts] flat

    float* z   = (float*)d_ws;                                        // 1 MiB
    int*   idx = (int*)((char*)d_ws + (size_t)NB * NPTS * sizeof(float)); // 256 KiB

    dim3 g1(NPTS / TILE_COLS, NB);            // (16, 16)
    colmax_kernel<<<g1, 256, 0, stream>>>(y, z);

    topk_sort_kernel<<<NB, 1024, (size_t)NPTS * sizeof(unsigned long long), stream>>>(z, idx);

    dim3 g3(CY / 8 + 1, NB);                  // (65, 16)
    gather_kernel<<<g3, 256, 0, stream>>>(x, y, idx, out);
}